// HierarchicalModel_14096082666274
// MI455X (gfx1250) — compile-verified
//
#include <hip/hip_runtime.h>

typedef __attribute__((ext_vector_type(2))) float v2f;
typedef __attribute__((ext_vector_type(8))) float v8f;

#define DIM 128

// Hierarchical-softmax path probability:
//   w = size_vertex + wo, L = bitlength(w), terms t = 0..L-3
//   node_t = w >> (L-2-t), sign_t = +1 iff bit (L-3-t) of w == 0
//   out = prod_t sigmoid(sign_t * dot(prob[node_t], phi[wi]))
//
// Implemented as a 32x128 @ 128 mat-vec via V_WMMA_F32_16X16X4_F32 (pure f32
// WMMA -> bit-compatible precision class with the f32 reference). Two 16-row
// tiles, K-loop of 32 steps of K=4; B is h broadcast across all 16 columns so
// every column of D carries the dot product.
//
// Padded rows (m >= nTerms) are clamped to prob row 0 and simply computed as
// garbage dots that the final reduction never reads — no masking VALU ops on
// the load->WMMA critical path.
__global__ __launch_bounds__(32)
void hs_wmma_kernel(const float* __restrict__ phi,
                    const float* __restrict__ prob,
                    const int* __restrict__ wi_p,
                    const int* __restrict__ wo_p,
                    float* __restrict__ out,
                    int size_vertex)
{
    __shared__ float s_dots[32];

    const int lane = threadIdx.x;               // one wave32
    const int wi = wi_p[0];
    const int wo = wo_p[0];
    const unsigned int w = (unsigned int)(size_vertex + wo);

    const int L = 32 - __clz((int)w);           // bit length of w (w >= 1)
    int nTerms = L - 2;
    if (nTerms < 0)  nTerms = 0;
    if (nTerms > 32) nTerms = 32;               // 32-row padding covers any int wo

    const float* __restrict__ h = phi + (size_t)wi * DIM;

    // A-matrix f32 16x4 layout: lanes 0-15 -> K=0,1 ; lanes 16-31 -> K=2,3.
    // B-matrix 4x16: same K split across half-waves; columns broadcast h.
    const int m    = lane & 15;
    const int koff = (lane < 16) ? 0 : 2;

    const int m0 = m;                           // tile0: rows 0..15
    const int m1 = m + 16;                      // tile1: rows 16..31
    // Clamp padded rows to row 0 (valid memory); their dots are never read.
    const unsigned int node0 = (m0 < nTerms) ? (w >> (L - 2 - m0)) : 0u;
    const unsigned int node1 = (m1 < nTerms) ? (w >> (L - 2 - m1)) : 0u;
    const float* __restrict__ row0 = prob + (size_t)node0 * DIM;
    const float* __restrict__ row1 = prob + (size_t)node1 * DIM;

    v8f acc0 = {};
    v8f acc1 = {};
    for (int kk = 0; kk < DIM; kk += 4) {
        v2f b  = *(const v2f*)(h    + kk + koff);      // same for all columns
        v2f a0 = *(const v2f*)(row0 + kk + koff);
        v2f a1 = *(const v2f*)(row1 + kk + koff);
        // 8 args: (neg_a, A, neg_b, B, c_mod, C, reuse_a, reuse_b)
        acc0 = __builtin_amdgcn_wmma_f32_16x16x4_f32(
            false, a0, false, b, (short)0, acc0, false, false);
        acc1 = __builtin_amdgcn_wmma_f32_16x16x4_f32(
            false, a1, false, b, (short)0, acc1, false, false);
    }

    // D layout: VGPR r -> M=r (lanes 0-15, N=lane) / M=8+r (lanes 16-31).
    // All 16 columns hold the same dot; read column 0 of each half-wave.
    if (lane == 0) {
        #pragma unroll
        for (int r = 0; r < 8; ++r) { s_dots[r]      = acc0[r]; s_dots[16 + r] = acc1[r]; }
    }
    if (lane == 16) {
        #pragma unroll
        for (int r = 0; r < 8; ++r) { s_dots[8 + r]  = acc0[r]; s_dots[24 + r] = acc1[r]; }
    }
    __syncthreads();

    if (lane == 0) {
        float p = 1.0f;
        for (int t = 0; t < nTerms; ++t) {
            float sgn = ((w >> (L - 3 - t)) & 1u) ? -1.0f : 1.0f;
            float d   = sgn * s_dots[t];
            p *= 1.0f / (1.0f + __expf(-d));           // sigmoid
        }
        out[0] = p;
    }
}

extern "C" void kernel_launch(void* const* d_in, const int* in_sizes, int n_in,
                              void* d_out, int out_size, void* d_ws, size_t ws_size,
                              hipStream_t stream) {
    const float* phi  = (const float*)d_in[0];   // [size_vertex, 128] f32
    const float* prob = (const float*)d_in[1];   // [2*size_vertex, 128] f32
    const int*   wi   = (const int*)d_in[2];     // scalar
    const int*   wo   = (const int*)d_in[3];     // scalar
    float*       out  = (float*)d_out;           // 1 element

    const int size_vertex = in_sizes[0] / DIM;

    hs_wmma_kernel<<<1, 32, 0, stream>>>(phi, prob, wi, wo, out, size_vertex);
}